// GeometricAffine_28243704939066
// MI455X (gfx1250) — compile-verified
//
#include <hip/hip_runtime.h>
#include <hip/hip_bf16.h>
#include <math.h>

typedef __attribute__((ext_vector_type(2))) float v2f;
typedef __attribute__((ext_vector_type(8))) float v8f;

#define BATCH   16
#define NPTS    2048
#define FPS_NUM 512
#define KNB     24
#define FEAT    64
#define BIGF    1e10f
#define CLEARED 1e30f
#define EPSF    1e-5f
#define CHUNK   512
#define CP      (CHUNK + 4)   // padded LDS stride: rows 0-7 vs 8-15 land on disjoint bank halves

// ---------------------------------------------------------------------------
// Kernel A: furthest point sampling. One block per batch, 1024 threads
// (32 wave32s). Points + running state in LDS/registers; 512 sequential
// argmax steps, each a shfl-xor wave reduce + 32-entry LDS reduce.
// Emits fps_idx (int32 bits) and sample_points directly into d_out.
// ---------------------------------------------------------------------------
__global__ __launch_bounds__(1024)
void fps_kernel(const float* __restrict__ points,
                int* __restrict__ fps_idx,
                float* __restrict__ sample_points) {
  __shared__ float px[NPTS], py[NPTS], pz[NPTS];
  __shared__ float swv[32];
  __shared__ int   swi[32];
  __shared__ float fxs, fys, fzs;
  __shared__ int   cfi;

  const int b   = blockIdx.x;
  const int tid = threadIdx.x;
  const int lane = tid & 31, wid = tid >> 5;
  const float* P = points + (size_t)b * NPTS * 3;

  for (int i = tid; i < NPTS; i += 1024) {
    px[i] = P[i * 3 + 0];
    py[i] = P[i * 3 + 1];
    pz[i] = P[i * 3 + 2];
  }
  if (tid == 0) cfi = 0;
  __syncthreads();
  if (tid == 0) { fxs = px[0]; fys = py[0]; fzs = pz[0]; }
  __syncthreads();

  const int i0 = tid, i1 = tid + 1024;
  float m0 = BIGF, m1 = BIGF;

  for (int it = 0; it < FPS_NUM; ++it) {
    const float fx = fxs, fy = fys, fz = fzs;
    const int cur = cfi;

    float dx = px[i0] - fx, dy = py[i0] - fy, dz = pz[i0] - fz;
    m0 = fminf(m0, dx * dx + dy * dy + dz * dz);
    dx = px[i1] - fx; dy = py[i1] - fy; dz = pz[i1] - fz;
    m1 = fminf(m1, dx * dx + dy * dy + dz * dz);

    float bv; int bi;
    if (m0 >= m1) { bv = m0; bi = i0; } else { bv = m1; bi = i1; }
    #pragma unroll
    for (int off = 16; off; off >>= 1) {
      float ov = __shfl_xor(bv, off, 32);
      int   oi = __shfl_xor(bi, off, 32);
      if (ov > bv || (ov == bv && oi < bi)) { bv = ov; bi = oi; }
    }
    if (lane == 0) { swv[wid] = bv; swi[wid] = bi; }
    __syncthreads();

    if (wid == 0) {
      float rv = swv[lane];
      int   ri = swi[lane];
      #pragma unroll
      for (int off = 16; off; off >>= 1) {
        float ov = __shfl_xor(rv, off, 32);
        int   oi = __shfl_xor(ri, off, 32);
        if (ov > rv || (ov == rv && oi < ri)) { rv = ov; ri = oi; }
      }
      if (lane == 0) {
        fps_idx[b * FPS_NUM + it] = cur;                      // emit current far
        float* sp = sample_points + (size_t)(b * FPS_NUM + it) * 3;
        sp[0] = fx; sp[1] = fy; sp[2] = fz;
        cfi = ri; fxs = px[ri]; fys = py[ri]; fzs = pz[ri];   // next far
      }
    }
    __syncthreads();
  }
}

// ---------------------------------------------------------------------------
// Kernel B: kNN via f32 WMMA (16x16x4) + exact top-24 selection.
// One block per (batch, 16-sample tile); 256 threads = 8 waves.
// A fragment = -2 * sample coords (K = x,y,z,0), B = point coords,
// so dist = |s|^2 + |p|^2 + C. Dist chunks of 512 points live in LDS;
// each wave does repeated wave-min extraction into a sorted 24-list.
// ---------------------------------------------------------------------------
__global__ __launch_bounds__(256)
void knn_kernel(const float* __restrict__ points,
                const float* __restrict__ sample_points,
                int* __restrict__ gidx_out) {
  __shared__ float px[NPTS], py[NPTS], pz[NPTS];
  __shared__ float distc[16 * CP];
  __shared__ float topd[16 * KNB];
  __shared__ int   topi[16 * KNB];
  __shared__ float sxh[16], syh[16], szh[16], ssh[16];

  const int b = blockIdx.y, stile = blockIdx.x;
  const int tid = threadIdx.x, lane = tid & 31, wid = tid >> 5;
  const float* P = points + (size_t)b * NPTS * 3;

  for (int i = tid; i < NPTS; i += 256) {
    px[i] = P[i * 3 + 0];
    py[i] = P[i * 3 + 1];
    pz[i] = P[i * 3 + 2];
  }
  if (tid < 16) {
    const float* sp = sample_points + (size_t)(b * FPS_NUM + stile * 16 + tid) * 3;
    float x = sp[0], y = sp[1], z = sp[2];
    sxh[tid] = x; syh[tid] = y; szh[tid] = z;
    ssh[tid] = x * x + y * y + z * z;
  }
  for (int t = tid; t < 16 * KNB; t += 256) { topd[t] = CLEARED; topi[t] = 0; }
  __syncthreads();

  // A fragment (16x4 f32): lanes 0-15 hold K=0,1 (x,y); lanes 16-31 hold K=2,3 (z,0)
  const bool hi = lane >= 16;
  const int  col = lane & 15;
  v2f a;
  a.x = -2.0f * (hi ? szh[col] : sxh[col]);
  a.y = hi ? 0.0f : (-2.0f * syh[col]);

  for (int c = 0; c < 4; ++c) {
    const int cb = c * CHUNK;
    // --- distance chunk: 32 tiles of 16 points, 4 tiles per wave ---
    #pragma unroll
    for (int j = 0; j < 4; ++j) {
      const int tloc = wid * 4 + j;
      const int pn = cb + tloc * 16 + col;
      const float x = px[pn], y = py[pn], z = pz[pn];
      v2f bm;
      bm.x = hi ? z : x;
      bm.y = hi ? 0.0f : y;
      const float pp = x * x + y * y + z * z;
      v8f acc = {0.f, 0.f, 0.f, 0.f, 0.f, 0.f, 0.f, 0.f};
      acc = __builtin_amdgcn_wmma_f32_16x16x4_f32(
          false, a, false, bm, (short)0, acc, false, false);
      const int rb = hi ? 8 : 0;
      #pragma unroll
      for (int v = 0; v < 8; ++v) {
        const int row = rb + v;
        distc[row * CP + tloc * 16 + col] = ssh[row] + pp + acc[v];
      }
    }
    __syncthreads();

    // --- top-24 merge: wave w owns rows 2w, 2w+1 ---
    for (int rr = 0; rr < 2; ++rr) {
      const int r = wid * 2 + rr;
      float v[16];
      #pragma unroll
      for (int j = 0; j < 16; ++j) v[j] = distc[r * CP + lane + 32 * j];

      for (int iter = 0; iter < KNB; ++iter) {
        float mv = v[0]; int mj = 0;
        #pragma unroll
        for (int j = 1; j < 16; ++j)
          if (v[j] < mv) { mv = v[j]; mj = j; }
        int mc = lane + 32 * mj;
        #pragma unroll
        for (int off = 16; off; off >>= 1) {
          float ov = __shfl_xor(mv, off, 32);
          int   oc = __shfl_xor(mc, off, 32);
          if (ov < mv || (ov == mv && oc < mc)) { mv = ov; mc = oc; }
        }
        const float tail = topd[r * KNB + KNB - 1];
        if (mv >= tail) break;            // nothing else in chunk can enter
        if (lane == 0) {
          int pos = KNB - 1;
          while (pos > 0 && topd[r * KNB + pos - 1] > mv) {
            topd[r * KNB + pos] = topd[r * KNB + pos - 1];
            topi[r * KNB + pos] = topi[r * KNB + pos - 1];
            --pos;
          }
          topd[r * KNB + pos] = mv;
          topi[r * KNB + pos] = cb + mc;
        }
        if ((mc & 31) == lane) v[mc >> 5] = CLEARED;  // retire candidate (registers)
      }
    }
    __syncthreads();
  }

  const int base = (b * FPS_NUM + stile * 16) * KNB;
  for (int t = tid; t < 16 * KNB; t += 256) gidx_out[base + t] = topi[t];
}

// ---------------------------------------------------------------------------
// Kernel C0/C1: batch-wide sum & sumsq of centered features (for ddof=1 std).
// ---------------------------------------------------------------------------
__global__ void zero_sums_kernel(float* sums) {
  if (threadIdx.x < 2 * BATCH) sums[threadIdx.x] = 0.0f;
}

__global__ __launch_bounds__(256)
void stats_kernel(const float* __restrict__ feats,
                  const int* __restrict__ fps_idx,
                  const int* __restrict__ gidx,
                  float* __restrict__ sums) {
  const int b = blockIdx.y, blk = blockIdx.x;       // 96 blocks of 128 (s,k) pairs
  const int tid = threadIdx.x;
  const int pair0 = blk * 128;
  float s = 0.0f, sq = 0.0f;
  for (int e = tid; e < 128 * FEAT; e += 256) {
    const int pr = pair0 + (e >> 6);
    const int f  = e & 63;
    const int sp = pr / KNB;
    const int gi = gidx[b * FPS_NUM * KNB + pr];
    const int si = fps_idx[b * FPS_NUM + sp];
    const float g = feats[((size_t)b * NPTS + gi) * FEAT + f];
    const float m = feats[((size_t)b * NPTS + si) * FEAT + f];
    const float cc = g - m;
    s += cc; sq += cc * cc;
  }
  #pragma unroll
  for (int off = 16; off; off >>= 1) {
    s  += __shfl_xor(s,  off, 32);
    sq += __shfl_xor(sq, off, 32);
  }
  __shared__ float ps[8], pq[8];
  const int lane = tid & 31, wid = tid >> 5;
  if (lane == 0) { ps[wid] = s; pq[wid] = sq; }
  __syncthreads();
  if (tid == 0) {
    float ts = 0.f, tq = 0.f;
    #pragma unroll
    for (int w = 0; w < 8; ++w) { ts += ps[w]; tq += pq[w]; }
    atomicAdd(&sums[b * 2 + 0], ts);   // global_atomic_add_f32
    atomicAdd(&sums[b * 2 + 1], tq);
  }
}

// ---------------------------------------------------------------------------
// Kernel C2: bandwidth-bound output stream (float4-vectorized, ~100 MB).
// out[b,s,k,0:64] = alpha*(g-m)/(std+eps)+beta ; out[b,s,k,64:128] = m
// ---------------------------------------------------------------------------
__global__ __launch_bounds__(256)
void out_kernel(const float* __restrict__ feats,
                const int* __restrict__ fps_idx,
                const int* __restrict__ gidx,
                const float* __restrict__ alpha,
                const float* __restrict__ beta,
                const float* __restrict__ sums,
                float* __restrict__ out_feats) {
  const int gid = blockIdx.x * 256 + threadIdx.x;
  const int row = gid >> 4;                 // (b,s,k) row, 16 threads per row
  const int f   = (gid & 15) * 4;
  const int b   = row / (FPS_NUM * KNB);
  const int rem = row - b * (FPS_NUM * KNB);
  const int sp  = rem / KNB;
  const int gi  = gidx[row];
  const int si  = fps_idx[b * FPS_NUM + sp];

  const float sum = sums[b * 2 + 0], sumsq = sums[b * 2 + 1];
  const float n = (float)(FPS_NUM * KNB * FEAT);
  const float var = (sumsq - sum * sum / n) / (n - 1.0f);
  const float invd = 1.0f / (sqrtf(var) + EPSF);

  const float4 g  = *(const float4*)(feats + ((size_t)b * NPTS + gi) * FEAT + f);
  const float4 m  = *(const float4*)(feats + ((size_t)b * NPTS + si) * FEAT + f);
  const float4 al = *(const float4*)(alpha + f);
  const float4 be = *(const float4*)(beta + f);

  float4 gf;
  gf.x = fmaf(al.x, (g.x - m.x) * invd, be.x);
  gf.y = fmaf(al.y, (g.y - m.y) * invd, be.y);
  gf.z = fmaf(al.z, (g.z - m.z) * invd, be.z);
  gf.w = fmaf(al.w, (g.w - m.w) * invd, be.w);

  float* ob = out_feats + (size_t)row * (2 * FEAT);
  *(float4*)(ob + f)        = gf;
  *(float4*)(ob + FEAT + f) = m;
}

// ---------------------------------------------------------------------------
extern "C" void kernel_launch(void* const* d_in, const int* in_sizes, int n_in,
                              void* d_out, int out_size, void* d_ws, size_t ws_size,
                              hipStream_t stream) {
  const float* points = (const float*)d_in[0];   // (16,2048,3)
  const float* feats  = (const float*)d_in[1];   // (16,2048,64)
  const float* alpha  = (const float*)d_in[2];   // (64)
  const float* beta   = (const float*)d_in[3];   // (64)

  float* out = (float*)d_out;
  int*   fps_idx_out   = (int*)out;                                   // 16*512 int32
  float* sample_points = out + BATCH * FPS_NUM;                       // 16*512*3
  float* out_feats     = out + BATCH * FPS_NUM + BATCH * FPS_NUM * 3; // 16*512*24*128

  int*   gidx = (int*)d_ws;                                           // 16*512*24
  float* sums = (float*)((char*)d_ws +
                         (size_t)BATCH * FPS_NUM * KNB * sizeof(int)); // 32 floats

  fps_kernel<<<BATCH, 1024, 0, stream>>>(points, fps_idx_out, sample_points);
  knn_kernel<<<dim3(FPS_NUM / 16, BATCH), 256, 0, stream>>>(points, sample_points, gidx);
  zero_sums_kernel<<<1, 64, 0, stream>>>(sums);
  stats_kernel<<<dim3(96, BATCH), 256, 0, stream>>>(feats, fps_idx_out, gidx, sums);
  const int total = BATCH * FPS_NUM * KNB * 16;   // threads: rows * 16
  out_kernel<<<total / 256, 256, 0, stream>>>(feats, fps_idx_out, gidx,
                                              alpha, beta, sums, out_feats);
}